// NoisedTopK_85968065397431
// MI455X (gfx1250) — compile-verified
//
#include <hip/hip_runtime.h>
#include <float.h>

// NoisedTopK: out[b] = mean_s( 5th-largest_c( x[b,c] + Z[b,c,s] ) )
// B=1024, C=32000, S=5, K=5, EPSILON=1.0 (so v = x + z).
// Memory-bound: 786 MB streamed once -> ~34 us floor at 23.3 TB/s.
// gfx1250 path: double-buffered GLOBAL_LOAD_ASYNC_TO_LDS_B128 staging with
// explicit S_WAIT_ASYNCCNT; falls back to direct b128 loads + prefetch.

#if defined(__HIP_DEVICE_COMPILE__) &&                                   \
    __has_builtin(__builtin_amdgcn_global_load_async_to_lds_b128) &&     \
    __has_builtin(__builtin_amdgcn_s_wait_asynccnt)
#define USE_ASYNC_LDS 1
#else
#define USE_ASYNC_LDS 0
#endif

namespace {

constexpr int kNS      = 5;      // noise samples
constexpr int kK       = 5;      // k-th largest
constexpr int kC       = 32000;  // class dim
constexpr int kThreads = 256;    // 8 wave32s per block
constexpr int kWaves   = kThreads / 32;
constexpr int kItems   = 4;                  // consecutive c per thread per tile
constexpr int kStride  = kThreads * kItems;  // 1024 c per tile
constexpr int kTiles   = (kC + kStride - 1) / kStride;  // 32 (last partial = 2 waves)
constexpr int kBytesPerThread = (kItems + kItems * kNS) * 4;  // 96 = 6 x 16B

#if USE_ASYNC_LDS
typedef int v4i_t __attribute__((ext_vector_type(4)));
typedef __attribute__((address_space(1))) v4i_t* gv4i_p;  // global v4i*
typedef __attribute__((address_space(3))) v4i_t* lv4i_p;  // LDS v4i*

__device__ __forceinline__ void asyncCopy16(const void* g, void* l) {
  __builtin_amdgcn_global_load_async_to_lds_b128(
      (gv4i_p)(void*)g, (lv4i_p)(void*)l, /*offset=*/0, /*cpol=*/0);
}
#endif

// Guarded sorted-descending insert into a 5-register list.
__device__ __forceinline__ void insert5(float (&t)[kK], float v) {
  if (v > t[kK - 1]) {
    t[kK - 1] = v;
#pragma unroll
    for (int k = kK - 1; k > 0; --k) {
      float a = t[k - 1], b = t[k];
      t[k - 1] = fmaxf(a, b);
      t[k]     = fminf(a, b);
    }
  }
}

__device__ __forceinline__ float waveMax32(float v) {
#pragma unroll
  for (int off = 16; off > 0; off >>= 1)
    v = fmaxf(v, __shfl_xor(v, off, 32));
  return v;
}

}  // namespace

__global__ __launch_bounds__(kThreads) void NoisedTopK_85968065397431_kernel(
    const float* __restrict__ x, const float* __restrict__ Z,
    float* __restrict__ out) {
  const int b    = blockIdx.x;
  const int tid  = threadIdx.x;
  const int lane = tid & 31;
  const int wave = tid >> 5;

  const float* __restrict__ xb = x + (size_t)b * kC;
  const float* __restrict__ Zb = Z + (size_t)b * kC * kNS;

  // Per-thread sorted top-5 per noise sample, all in VGPRs (25 regs).
  float top[kNS][kK];
#pragma unroll
  for (int s = 0; s < kNS; ++s)
#pragma unroll
    for (int k = 0; k < kK; ++k) top[s][k] = -FLT_MAX;

#if USE_ASYNC_LDS
  // Double-buffered per-thread private staging: 2 x 256 x 96B = 48KB LDS.
  __shared__ __align__(16) char lbuf[2][kThreads][kBytesPerThread];

  // Issue 6 async b128 copies (16B x + 80B Z) into this thread's LDS slot.
  auto issueTile = [&](int t) {
    const int c0 = t * kStride + tid * kItems;
    if (c0 < kC) {  // EXEC-masked lanes copy nothing (wave-uniform cut)
      char* l = &lbuf[t & 1][tid][0];
      const char* gx = (const char*)(xb + c0);
      const char* gz = (const char*)(Zb + (size_t)c0 * kNS);
      asyncCopy16(gx, l);
#pragma unroll
      for (int q = 0; q < kItems * kNS / 4; ++q)
        asyncCopy16(gz + 16 * q, l + 16 + 16 * q);
    }
  };

  auto consumeTile = [&](int t) {
    const int c0 = t * kStride + tid * kItems;
    if (c0 < kC) {
      const char* l = &lbuf[t & 1][tid][0];
      const float4 xv = *reinterpret_cast<const float4*>(l);
      float zr[kItems * kNS];
#pragma unroll
      for (int q = 0; q < kItems * kNS / 4; ++q)
        reinterpret_cast<float4*>(zr)[q] =
            *reinterpret_cast<const float4*>(l + 16 + 16 * q);
      const float xa[kItems] = {xv.x, xv.y, xv.z, xv.w};
#pragma unroll
      for (int i = 0; i < kItems; ++i)
#pragma unroll
        for (int s = 0; s < kNS; ++s)
          insert5(top[s], xa[i] + zr[i * kNS + s]);
    }
  };

  issueTile(0);
  for (int t = 0; t < kTiles; ++t) {
    // Wave-uniform: does this wave issue 6 async ops for tile t+1?
    const int cNextWave = (t + 1) * kStride + (tid & ~31) * kItems;
    if ((t + 1 < kTiles) && (cNextWave < kC)) {
      issueTile(t + 1);
      __builtin_amdgcn_s_wait_asynccnt(6);  // tile t's 6 ops complete (in-order)
    } else {
      __builtin_amdgcn_s_wait_asynccnt(0);  // drain before final tile consume
    }
    consumeTile(t);
  }
#else
  // Fallback: direct b128 streaming + gfx1250 global_prefetch_b8.
  for (int c0 = tid * kItems; c0 < kC; c0 += kStride) {
    const int cn = c0 + kStride;
    if (cn < kC) {
      __builtin_prefetch(Zb + (size_t)cn * kNS, 0, 0);
      __builtin_prefetch(xb + cn, 0, 0);
    }
    const float4 xv = *reinterpret_cast<const float4*>(xb + c0);
    float zr[kItems * kNS];
    const float4* zsrc =
        reinterpret_cast<const float4*>(Zb + (size_t)c0 * kNS);
#pragma unroll
    for (int q = 0; q < (kItems * kNS) / 4; ++q)
      reinterpret_cast<float4*>(zr)[q] = zsrc[q];
    const float xa[kItems] = {xv.x, xv.y, xv.z, xv.w};
#pragma unroll
    for (int i = 0; i < kItems; ++i)
#pragma unroll
      for (int s = 0; s < kNS; ++s)
        insert5(top[s], xa[i] + zr[i * kNS + s]);
  }
#endif

  // Wave-level exact top-5 per s: 5 rounds of "pop the wave max".
  __shared__ float sc[kWaves][kNS][kK];
  __shared__ float skth[kNS];

#pragma unroll
  for (int s = 0; s < kNS; ++s) {
    int ptr = 0;  // per-lane cursor into its sorted list
#pragma unroll
    for (int k = 0; k < kK; ++k) {
      float cur = (ptr == 0)   ? top[s][0]
                  : (ptr == 1) ? top[s][1]
                  : (ptr == 2) ? top[s][2]
                  : (ptr == 3) ? top[s][3]
                  : (ptr == 4) ? top[s][4]
                               : -FLT_MAX;
      float m = waveMax32(cur);
      unsigned long long bal = __ballot(cur == m);  // wave32: low 32 bits
      int first = (int)(__ffsll(bal) - 1);
      if (lane == first) ++ptr;  // exactly one lane pops
      if (lane == 0) sc[wave][s][k] = m;
    }
  }
  __syncthreads();

  // Block merge: 8 waves x 5 candidates per s -> 5th largest of 40.
  if (tid < kNS) {
    float t[kK];
#pragma unroll
    for (int k = 0; k < kK; ++k) t[k] = -FLT_MAX;
    for (int w = 0; w < kWaves; ++w)
#pragma unroll
      for (int k = 0; k < kK; ++k) insert5(t, sc[w][tid][k]);
    skth[tid] = t[kK - 1];
  }
  __syncthreads();

  if (tid == 0) {
    float acc = 0.0f;
#pragma unroll
    for (int s = 0; s < kNS; ++s) acc += skth[s];
    out[b] = acc * (1.0f / kNS);
  }
}

extern "C" void kernel_launch(void* const* d_in, const int* in_sizes, int n_in,
                              void* d_out, int out_size, void* d_ws,
                              size_t ws_size, hipStream_t stream) {
  const float* x = (const float*)d_in[0];  // [B, C] f32
  const float* Z = (const float*)d_in[1];  // [B, C, S] f32
  float* out = (float*)d_out;              // [B] f32

  dim3 grid(out_size > 0 ? out_size : 1);  // one block per b
  dim3 block(kThreads);
  hipLaunchKernelGGL(NoisedTopK_85968065397431_kernel, grid, block, 0, stream,
                     x, Z, out);
}